// SequenceSmoothLossCtc_v11_31748398252076
// MI455X (gfx1250) — compile-verified
//
#include <hip/hip_runtime.h>
#include <math.h>

// Problem constants (match reference)
#define Bsz   128
#define Tt    64
#define Cc    6625
#define Ll    25
#define Kk    6
#define LSs   25
#define NCLS  (1 + Ll + Kk * LSs)   // 176 gathered classes per sample
#define Ssz   (2 * Ll + 1)          // 51 CTC lattice states
#define NEGF  (-1e30f)

typedef __attribute__((ext_vector_type(2))) float v2f;
typedef __attribute__((ext_vector_type(8))) float v8f;

// fast log-add-exp on hardware transcendentals (hot paths)
__device__ __forceinline__ float lae2(float a, float b) {
    float m = fmaxf(a, b);
    if (m < -1e29f) return NEGF;
    return m + __logf(__expf(a - m) + __expf(b - m));
}

// online-softmax + top-2 single-element update
__device__ __forceinline__ void upd(float x, int c, float& m, float& s,
                                    float& v1, int& i1, float& v2, int& i2) {
    if (x > m) { s = s * __expf(m - x) + 1.0f; m = x; }
    else       { s += __expf(x - m); }
    if (c != 0) {                                // top-2 excludes blank
        if (x > v1)      { v2 = v1; i2 = i1; v1 = x; i1 = c; }
        else if (x > v2) { v2 = x;  i2 = c; }
    }
}

__device__ __forceinline__ void merge2(float ov, int oi, float& v1, int& i1,
                                       float& v2, int& i2) {
    if (ov > v1)      { v2 = v1; i2 = i1; v1 = ov; i1 = oi; }
    else if (ov > v2) { v2 = ov; i2 = oi; }
}

// ---------------------------------------------------------------------------
// Kernel 0: build per-sample class list: [blank, text[b,:], smooth_text[b,:,:]]
// ---------------------------------------------------------------------------
__global__ void k_build_cls(const int* __restrict__ text,
                            const int* __restrict__ smooth_text,
                            int* __restrict__ cls) {
    int idx = blockIdx.x * blockDim.x + threadIdx.x;
    if (idx >= Bsz * NCLS) return;
    int b = idx / NCLS, j = idx % NCLS;
    int v;
    if (j == 0)          v = 0;                                  // blank
    else if (j <= Ll)    v = text[b * Ll + (j - 1)];             // master labels
    else                 v = smooth_text[b * Kk * LSs + (j - 1 - Ll)];
    cls[idx] = v;
}

// ---------------------------------------------------------------------------
// Kernel 1: fused streaming pass over preds row (b,t):
//   float4 streaming, online softmax (max + sum-exp), top-2 over c>=1,
//   gather of the 176 needed lp values while the row is cache-hot.
//   Per-wave exp-sum reduction via V_WMMA_F32_16X16X4_F32 (ones-B).
// ---------------------------------------------------------------------------
__global__ __launch_bounds__(256) void k_stats(
    const float* __restrict__ preds, const int* __restrict__ cls,
    float* __restrict__ logZ, float* __restrict__ t1v, float* __restrict__ t2v,
    int* __restrict__ t1i, int* __restrict__ t2i, float* __restrict__ lp_tab) {
    __shared__ float wmax[8], w1v[8], w2v[8], wsum[8];
    __shared__ int   w1i[8], w2i[8];
    __shared__ float sM, sLogZ;

    const int bt  = blockIdx.x;          // b*T + t
    const int b   = bt / Tt;
    const int tid = threadIdx.x;
    const int lane = tid & 31, wid = tid >> 5;
    const float* row = preds + (size_t)bt * Cc;

    float m = -INFINITY, s = 0.f;
    float v1 = NEGF, v2 = NEGF;
    int   i1 = 0,    i2 = 0;

    // row base element index is bt*6625 ≡ bt (mod 4): 0..3-elt prologue aligns
    const int p = (4 - (bt & 3)) & 3;
    if (tid < p) upd(row[tid], tid, m, s, v1, i1, v2, i2);

    const float4* rp = reinterpret_cast<const float4*>(row + p);
    const int nmain4 = (Cc - p) >> 2;
    for (int i = tid; i < nmain4; i += 256) {
        __builtin_prefetch(rp + i + 512, 0, 1);     // global_prefetch_b8
        float4 x4 = rp[i];
        int c = p + (i << 2);
        upd(x4.x, c + 0, m, s, v1, i1, v2, i2);
        upd(x4.y, c + 1, m, s, v1, i1, v2, i2);
        upd(x4.z, c + 2, m, s, v1, i1, v2, i2);
        upd(x4.w, c + 3, m, s, v1, i1, v2, i2);
    }
    int ctail = p + (nmain4 << 2) + tid;
    if (ctail < Cc) upd(row[ctail], ctail, m, s, v1, i1, v2, i2);

    // ---- wave-level reductions ----
    float om = m;                                   // keep own max for rescale
    float wm = m;
    for (int off = 16; off > 0; off >>= 1)
        wm = fmaxf(wm, __shfl_xor(wm, off, 32));
    for (int off = 16; off > 0; off >>= 1) {
        float ov1 = __shfl_xor(v1, off, 32); int oi1 = __shfl_xor(i1, off, 32);
        float ov2 = __shfl_xor(v2, off, 32); int oi2 = __shfl_xor(i2, off, 32);
        merge2(ov1, oi1, v1, i1, v2, i2);
        merge2(ov2, oi2, v1, i1, v2, i2);
    }
    if (lane == 0) {
        wmax[wid] = wm;
        w1v[wid] = v1; w1i[wid] = i1; w2v[wid] = v2; w2i[wid] = i2;
    }
    __syncthreads();

    if (tid == 0) {
        float M = wmax[0];
        for (int w = 1; w < 8; ++w) M = fmaxf(M, wmax[w]);
        sM = M;
    }
    __syncthreads();

    // rescaled per-thread partial; wave-sum via WMMA against an all-ones B:
    // D[m,n] = sum_k A[m,k]; per-lane sum of 8 D regs + xor-16 = full wave sum.
    float e = __expf(om - sM) * s;
    v2f a;    a.x = e;    a.y = 0.0f;
    v2f ones; ones.x = 1.0f; ones.y = 1.0f;
    v8f acc = {};
    acc = __builtin_amdgcn_wmma_f32_16x16x4_f32(false, a, false, ones,
                                                (short)0, acc, false, false);
    float wv = acc[0] + acc[1] + acc[2] + acc[3] +
               acc[4] + acc[5] + acc[6] + acc[7];
    wv += __shfl_xor(wv, 16, 32);
    if (lane == 0) wsum[wid] = wv;
    __syncthreads();

    if (tid == 0) {
        float S = 0.f;
        for (int w = 0; w < 8; ++w) S += wsum[w];
        float lz = sM + logf(S);                    // precise, once per block
        sLogZ = lz;
        logZ[bt] = lz;
        float T1 = NEGF, T2 = NEGF; int I1 = 0, I2 = 0;
        for (int w = 0; w < 8; ++w) {
            merge2(w1v[w], w1i[w], T1, I1, T2, I2);
            merge2(w2v[w], w2i[w], T1, I1, T2, I2);
        }
        t1v[bt] = T1 - lz; t2v[bt] = T2 - lz; t1i[bt] = I1; t2i[bt] = I2;
    }
    __syncthreads();

    // gather the 176 needed classes while the row is cache-hot
    float lz = sLogZ;
    for (int j = tid; j < NCLS; j += 256) {
        int cidx = cls[b * NCLS + j];
        lp_tab[(size_t)bt * NCLS + j] = row[cidx] - lz;
    }
}

// ---------------------------------------------------------------------------
// Kernel 2: beam-size-1 CTC prefix search confidence (serial over T per sample)
// ---------------------------------------------------------------------------
__global__ void k_beam(const float* __restrict__ preds,
                       const float* __restrict__ logZ,
                       const float* __restrict__ lp_tab,
                       const float* __restrict__ t1v, const float* __restrict__ t2v,
                       const int* __restrict__ t1i, const int* __restrict__ t2i,
                       float* __restrict__ conf) {
    int b = blockIdx.x * blockDim.x + threadIdx.x;
    if (b >= Bsz) return;
    float lpb = 0.f, lpnb = NEGF;
    int last = -1;
    for (int t = 0; t < Tt; ++t) {
        int bt = b * Tt + t;
        float lz     = logZ[bt];
        float lpt0   = lp_tab[(size_t)bt * NCLS];    // blank
        float tot    = lae2(lpb, lpnb);
        float new_pb = tot + lpt0;
        float rep = NEGF;
        if (last >= 0) {
            float lp_last = preds[(size_t)bt * Cc + last] - lz;
            rep = lpnb + lp_last;
        }
        float keep = lae2(new_pb, rep);
        float a1vv = t1v[bt], a2vv = t2v[bt];
        int   a1   = t1i[bt], a2   = t2i[bt];
        float best_ext; int best_c;
        if (last == a1) {   // best extension is lpb+top1 vs tot+top2
            float x = lpb + a1vv, y = tot + a2vv;
            if (x >= y) { best_ext = x; best_c = a1; }
            else        { best_ext = y; best_c = a2; }
        } else {            // ext[last] <= tot+top1, so top1 wins
            best_ext = tot + a1vv; best_c = a1;
        }
        bool take = best_ext > keep;
        lpb  = take ? NEGF     : new_pb;
        lpnb = take ? best_ext : rep;
        last = take ? best_c   : last;
    }
    conf[b] = expf(lae2(lpb, lpnb) / (float)Tt);    // precise, once per sample
}

// ---------------------------------------------------------------------------
// Kernel 3: CTC alpha recursion; one block per target (896), S=51 states in LDS
// ---------------------------------------------------------------------------
__global__ __launch_bounds__(64) void k_ctc(
    const float* __restrict__ lp_tab,
    const int* __restrict__ text, const int* __restrict__ smooth_text,
    const int* __restrict__ preds_size, const int* __restrict__ length,
    const int* __restrict__ smooth_length,
    float* __restrict__ nll_m, float* __restrict__ nll_s) {
    __shared__ float alpha[Ssz];
    const int n = blockIdx.x;
    const bool master = (n < Bsz);
    int b, base_j, tlen, in_len;
    const int* labs;
    if (master) {
        b = n; labs = text + b * Ll; base_j = 1;
        tlen = length[b]; in_len = preds_size[b];
    } else {
        int mm = n - Bsz; b = mm / Kk; int k = mm % Kk;
        labs = smooth_text + (size_t)mm * LSs;
        base_j = 1 + Ll + k * LSs;
        tlen = smooth_length[mm]; in_len = Tt;
    }
    const int s = threadIdx.x;
    int myj = 0; bool allow = false;
    if (s < Ssz) {
        if (s & 1) {
            int pos = (s - 1) >> 1;
            myj = base_j + pos;
            int lab = labs[pos];
            allow = (lab != 0) && (s == 1 || lab != labs[pos - 1]);
        }
        const float* lp0 = lp_tab + (size_t)(b * Tt) * NCLS;
        alpha[s] = (s == 0) ? lp0[0] : (s == 1 ? lp0[base_j] : NEGF);
    }
    __syncthreads();
    for (int t = 1; t < Tt; ++t) {
        float val = NEGF;
        if (s < Ssz) {
            float a  = alpha[s];
            float a1 = (s >= 1) ? alpha[s - 1] : NEGF;
            float a2 = (allow && s >= 2) ? alpha[s - 2] : NEGF;
            float m = fmaxf(a, fmaxf(a1, a2));
            float tot = (m < -1e29f) ? NEGF
                      : m + __logf(__expf(a - m) + __expf(a1 - m) + __expf(a2 - m));
            float lp = lp_tab[(size_t)(b * Tt + t) * NCLS + myj];
            float nv = tot + lp;
            val = (t < in_len) ? nv : a;
        }
        __syncthreads();
        if (s < Ssz) alpha[s] = val;
        __syncthreads();
    }
    if (s == 0) {
        int il = 2 * tlen;
        float ab = alpha[il];
        float ac = alpha[(il > 0) ? (il - 1) : 0];
        float nll = -lae2(ab, ac);
        if (nll > 1e29f) nll = 0.f;                 // zero_infinity
        if (master) nll_m[n] = nll; else nll_s[n - Bsz] = nll;
    }
}

// ---------------------------------------------------------------------------
// Kernel 4: combine into the final scalar loss
// ---------------------------------------------------------------------------
__global__ void k_final(const float* __restrict__ nll_m,
                        const float* __restrict__ nll_s,
                        const float* __restrict__ conf,
                        const int* __restrict__ length,
                        const int* __restrict__ smooth_length,
                        float* __restrict__ out) {
    __shared__ float red[Bsz];
    int b = threadIdx.x;
    float v = 0.f;
    if (b < Bsz) {
        float lm = nll_m[b] / (float)length[b];
        float cal = 0.f;
        for (int k = 0; k < Kk; ++k)
            cal += nll_s[b * Kk + k] / (float)smooth_length[b * Kk + k];
        cal /= (float)Kk;
        float c = conf[b];
        float ranking = 0.01f + 0.99f * (1.f - c) * (1.f - c);  // SMOOTH_TAIL
        v = lm + 0.1f * ranking * cal;                           // ALPHA
    }
    if (b < Bsz) red[b] = v;
    __syncthreads();
    if (b == 0) {
        float sres = 0.f;
        for (int i = 0; i < Bsz; ++i) sres += red[i];
        out[0] = sres / (float)Bsz;
    }
}

// ---------------------------------------------------------------------------
extern "C" void kernel_launch(void* const* d_in, const int* in_sizes, int n_in,
                              void* d_out, int out_size, void* d_ws, size_t ws_size,
                              hipStream_t stream) {
    const float* preds         = (const float*)d_in[0];
    const int*   text          = (const int*)  d_in[1];
    const int*   preds_size    = (const int*)  d_in[2];
    const int*   length        = (const int*)  d_in[3];
    const int*   smooth_text   = (const int*)  d_in[4];
    const int*   smooth_length = (const int*)  d_in[5];
    float*       out           = (float*)d_out;

    // workspace layout (~6.03 MB total)
    const int BT = Bsz * Tt;
    float* logZ   = (float*)d_ws;
    float* t1v    = logZ + BT;
    float* t2v    = t1v + BT;
    int*   t1i    = (int*)(t2v + BT);
    int*   t2i    = t1i + BT;
    int*   cls    = t2i + BT;                       // Bsz*NCLS ints
    float* lp_tab = (float*)(cls + Bsz * NCLS);     // BT*NCLS floats
    float* conf   = lp_tab + (size_t)BT * NCLS;
    float* nll_m  = conf + Bsz;
    float* nll_s  = nll_m + Bsz;

    k_build_cls<<<(Bsz * NCLS + 255) / 256, 256, 0, stream>>>(text, smooth_text, cls);
    k_stats<<<BT, 256, 0, stream>>>(preds, cls, logZ, t1v, t2v, t1i, t2i, lp_tab);
    k_beam<<<1, 128, 0, stream>>>(preds, logZ, lp_tab, t1v, t2v, t1i, t2i, conf);
    k_ctc<<<Bsz + Bsz * Kk, 64, 0, stream>>>(lp_tab, text, smooth_text, preds_size,
                                             length, smooth_length, nll_m, nll_s);
    k_final<<<1, 128, 0, stream>>>(nll_m, nll_s, conf, length, smooth_length, out);
}